// PredictionAggregator_36172214567691
// MI455X (gfx1250) — compile-verified
//
#include <hip/hip_runtime.h>
#include <math.h>

// Problem constants from the reference
#define B_  8
#define G_  2048
#define P_  64
#define F_  36
#define FP  48            // features padded to 3 WMMA column tiles
#define VOXEL 0.05f

typedef __attribute__((ext_vector_type(2))) float v2f;
typedef __attribute__((ext_vector_type(8))) float v8f;

__global__ void pa_zero_kernel(float* __restrict__ out, int n) {
    int i = blockIdx.x * blockDim.x + threadIdx.x;
    int stride = gridDim.x * blockDim.x;
    for (; i < n; i += stride) out[i] = 0.0f;
}

// One wave32 per (b,g) group.
__launch_bounds__(32)
__global__ void pa_agg_kernel(const float* __restrict__ pred,
                              const unsigned char* __restrict__ active_mask,
                              float* __restrict__ out, int nmax)
{
    __shared__ float Xp[P_ * FP];                 // 64x48 padded features (12 KB)
    __shared__ float W[P_ * P_];                  // 64x64 scatter-weight matrix (16 KB)
    __shared__ unsigned long long keys[P_];       // packed voxel keys
    __shared__ int   firstf[P_];                  // first-occurrence flags
    __shared__ float probs[P_];                   // sigmoid(conf)
    __shared__ int   nsegLDS;

    const int gidx = blockIdx.x;                  // b*G + g
    const int lane = threadIdx.x;                 // 0..31 (wave32)

    if (!active_mask[gidx]) return;               // uniform per block

    const float* __restrict__ src = pred + (size_t)gidx * (P_ * F_);
    // Early prefetch of this group's 9216B block (global_prefetch_b8)
    __builtin_prefetch(src,        0, 0);
    __builtin_prefetch(src +  512, 0, 0);
    __builtin_prefetch(src + 1024, 0, 0);
    __builtin_prefetch(src + 1536, 0, 0);
    __builtin_prefetch(src + 2048, 0, 0);

    // ---- Stage 64x36 block into zero-padded 64x48 LDS (coalesced) ----
    for (int e = lane; e < P_ * F_; e += 32) {
        int p = e / F_, f = e % F_;
        Xp[p * FP + f] = src[e];
    }
    for (int e = lane; e < P_ * (FP - F_); e += 32) {
        int p = e / (FP - F_), f = F_ + e % (FP - F_);
        Xp[p * FP + f] = 0.0f;
    }
    __syncthreads();

    // ---- Voxel keys + sigmoid(conf); 2 points per lane ----
    for (int t = 0; t < 2; ++t) {
        int p = lane + 32 * t;
        long long vx = (long long)floorf(Xp[p * FP + 0] / VOXEL) + (1 << 20);
        long long vy = (long long)floorf(Xp[p * FP + 1] / VOXEL) + (1 << 20);
        long long vz = (long long)floorf(Xp[p * FP + 2] / VOXEL) + (1 << 20);
        keys[p] = ((unsigned long long)(vx & 0x1FFFFFll) << 42)
                | ((unsigned long long)(vy & 0x1FFFFFll) << 21)
                |  (unsigned long long)(vz & 0x1FFFFFll);
        float lg = Xp[p * FP + 3];
        probs[p] = 1.0f / (1.0f + __expf(-lg));
    }
    __syncthreads();

    // ---- First-occurrence flags; also zero the scatter matrix ----
    for (int t = 0; t < 2; ++t) {
        int p = lane + 32 * t;
        unsigned long long k = keys[p];
        int first = 1;
        for (int j = 0; j < p; ++j) first &= (keys[j] != k);
        firstf[p] = first;
    }
    for (int e = lane; e < P_ * P_; e += 32) W[e] = 0.0f;
    __syncthreads();

    // ---- Rank (= #distinct smaller keys -> sorted-unique slot) + weights ----
    for (int t = 0; t < 2; ++t) {
        int p = lane + 32 * t;
        unsigned long long k = keys[p];
        int r = 0; float s = 0.0f;
        for (int j = 0; j < P_; ++j) {
            unsigned long long kj = keys[j];
            r += (firstf[j] && (kj < k)) ? 1 : 0;
            s += (kj == k) ? probs[j] : 0.0f;
        }
        W[r * P_ + p] = probs[p] / s;
    }
    if (lane == 0) {
        int c = 0;
        for (int j = 0; j < P_; ++j) c += firstf[j];
        nsegLDS = c;
    }
    __syncthreads();
    const int nseg = nsegLDS;

    // ---- GEMM: out(64seg x 48) = W(64x64) * Xp(64x48) via V_WMMA_F32_16X16X4_F32 ----
    const int half = lane >> 4;       // 0: lanes 0-15, 1: lanes 16-31
    const int l16  = lane & 15;
    const size_t outbase = (size_t)gidx * (size_t)nmax * F_;

    for (int ti = 0; ti < 4; ++ti) {
        for (int tj = 0; tj < 3; ++tj) {
            v8f acc = {0.f, 0.f, 0.f, 0.f, 0.f, 0.f, 0.f, 0.f};
            for (int k0 = 0; k0 < P_; k0 += 4) {
                // A: 16x4 f32. VGPR0: K=0 (lanes0-15) / K=2 (lanes16-31); VGPR1: K=1 / K=3
                const int ka = k0 + half * 2;
                v2f a, b;
                a.x = W[(ti * 16 + l16) * P_ + ka];
                a.y = W[(ti * 16 + l16) * P_ + ka + 1];
                // B: 4x16 f32, row-striped across lanes within a VGPR
                b.x = Xp[ka       * FP + tj * 16 + l16];
                b.y = Xp[(ka + 1) * FP + tj * 16 + l16];
                acc = __builtin_amdgcn_wmma_f32_16x16x4_f32(
                    /*neg_a=*/false, a, /*neg_b=*/false, b,
                    /*c_mod=*/(short)0, acc, /*reuse_a=*/false, /*reuse_b=*/false);
            }
            // D 16x16 f32: VGPR r -> M = r (lanes 0-15) / r+8 (lanes 16-31), N = lane%16
            for (int r = 0; r < 8; ++r) {
                int row = ti * 16 + r + half * 8;
                int col = tj * 16 + l16;
                if (row < nseg && col < F_) {
                    float v = acc[r];
                    if (col == 3) v = 1.0f / (1.0f + __expf(-v));  // agg conf -> prob
                    out[outbase + (size_t)row * F_ + col] = v;
                }
            }
        }
    }
}

extern "C" void kernel_launch(void* const* d_in, const int* in_sizes, int n_in,
                              void* d_out, int out_size, void* d_ws, size_t ws_size,
                              hipStream_t stream) {
    (void)n_in; (void)d_ws; (void)ws_size;
    const float* pred = (const float*)d_in[0];
    // active_mask: bernoulli/bool -> 1 byte per element (numpy bool layout)
    const unsigned char* mask = (const unsigned char*)d_in[1];
    float* out = (float*)d_out;

    // N_max is data-dependent in the reference; recover it from the output size.
    int nmax = out_size / (B_ * G_ * F_);
    if (nmax <= 0) nmax = 1;

    // Output is sparse-scattered: pre-zero it every call (harness poisons d_out).
    pa_zero_kernel<<<2048, 256, 0, stream>>>(out, out_size);
    // One wave32 per (b,g) group.
    pa_agg_kernel<<<B_ * G_, 32, 0, stream>>>(pred, mask, out, nmax);
}